// QKVProjectedLinear_27384711479893
// MI455X (gfx1250) — compile-verified
//
#include <hip/hip_runtime.h>
#include <hip/hip_bf16.h>

// Problem constants
#define D_LARGE 1024
#define D_SMALL 768
#define RANK    64
#define NROWS   16384      // BATCH*SEQ = 2*8192
#define OUTW    3072       // 3*D_LARGE

typedef __attribute__((ext_vector_type(2))) float v2f;
typedef __attribute__((ext_vector_type(8))) float v8f;

// ---------------------------------------------------------------------------
// Phase 1a: T[k][r][s] = sum_t B[k][r][t] * Ws[k][s][t]     (B * Ws^T, 64x768)
// ---------------------------------------------------------------------------
__global__ void __launch_bounds__(256)
k_precompute_T(const float* __restrict__ Bf, const float* __restrict__ Ws,
               float* __restrict__ T) {
    int tid = blockIdx.x * blockDim.x + threadIdx.x;
    if (tid >= 3 * 64 * 768) return;
    int k = tid / (64 * 768);
    int r = (tid / 768) % 64;
    int s = tid % 768;
    const float* brow = Bf + ((size_t)k * 64 + r) * 768;
    const float* wrow = Ws + ((size_t)k * 768 + s) * 768;
    float acc = 0.f;
    for (int t = 0; t < 768; ++t) acc = fmaf(brow[t], wrow[t], acc);
    T[tid] = acc;
}

// ---------------------------------------------------------------------------
// Phase 1b: M[k][r][rp] = sum_s T[k][r][s] * B[k][rp][s]    (T * B^T, 64x64)
// ---------------------------------------------------------------------------
__global__ void __launch_bounds__(256)
k_precompute_M(const float* __restrict__ T, const float* __restrict__ Bf,
               float* __restrict__ M) {
    int tid = blockIdx.x * blockDim.x + threadIdx.x;
    if (tid >= 3 * 64 * 64) return;
    int k  = tid / (64 * 64);
    int r  = (tid / 64) % 64;
    int rp = tid % 64;
    const float* trow = T  + ((size_t)k * 64 + r)  * 768;
    const float* brow = Bf + ((size_t)k * 64 + rp) * 768;
    float acc = 0.f;
    for (int s = 0; s < 768; ++s) acc = fmaf(trow[s], brow[s], acc);
    M[tid] = acc;
}

// ---------------------------------------------------------------------------
// Phase 1c: A2T[k][rp][d] = sum_r A[k][d][r] * M[k][r][rp]  ((A*M)^T, 64x1024)
// Stored transposed so phase-2 B-fragments are contiguous float2 loads.
// ---------------------------------------------------------------------------
__global__ void __launch_bounds__(256)
k_precompute_A2T(const float* __restrict__ A, const float* __restrict__ M,
                 float* __restrict__ A2T) {
    int tid = blockIdx.x * blockDim.x + threadIdx.x;
    if (tid >= 3 * 64 * 1024) return;
    int k  = tid / (64 * 1024);
    int rp = (tid / 1024) % 64;
    int d  = tid % 1024;
    const float* arow = A + ((size_t)k * 1024 + d) * 64;   // A[k][d][*]
    const float* mk   = M + (size_t)k * 64 * 64;
    float acc = 0.f;
    for (int r = 0; r < 64; ++r) acc = fmaf(arow[r], mk[r * 64 + rp], acc);
    A2T[tid] = acc;   // tid == (k*64 + rp)*1024 + d
}

// ---------------------------------------------------------------------------
// Phase 2: U[row][k*64+rp] = sum_d x[row][d] * A2[k][d][rp]
// GEMM M=16384, N=64 (per k), K=1024 using V_WMMA_F32_16X16X4_F32.
// Block = 256 threads = 8 waves; each wave: 16 rows x 64 cols, 4 accumulators.
// ---------------------------------------------------------------------------
__global__ void __launch_bounds__(256)
k_phase2_xA2(const float* __restrict__ x, const float* __restrict__ A2T,
             float* __restrict__ U) {
    const int lane  = threadIdx.x & 31;
    const int wave  = threadIdx.x >> 5;
    const int m     = lane & 15;     // row/col within 16 for this lane
    const int khalf = lane >> 4;     // 0: K={0,1}, 1: K={2,3} per ISA layout
    const int row0  = blockIdx.x * 128 + wave * 16;
    const int k     = blockIdx.y;

    const float* xrow = x   + (size_t)(row0 + m) * D_LARGE;
    const float* a2k  = A2T + (size_t)k * 64 * 1024;        // [rp][d]

    v8f acc[4] = {v8f{0}, v8f{0}, v8f{0}, v8f{0}};

    for (int kt = 0; kt < 256; ++kt) {
        const int kb = kt * 4 + khalf * 2;
        // A fragment: x rows, contiguous K pair
        v2f a = *(const v2f*)(xrow + kb);
#pragma unroll
        for (int j = 0; j < 4; ++j) {
            const int rp = j * 16 + m;
            // B fragment: A2T[rp][kb], A2T[rp][kb+1] contiguous
            v2f b = *(const v2f*)(a2k + (size_t)rp * 1024 + kb);
            acc[j] = __builtin_amdgcn_wmma_f32_16x16x4_f32(
                false, a, false, b, (short)0, acc[j], false, false);
        }
    }

    // D layout: VGPR v -> row m0+v (lanes 0-15) / m0+8+v (lanes 16-31), col = m
#pragma unroll
    for (int j = 0; j < 4; ++j) {
#pragma unroll
        for (int v = 0; v < 8; ++v) {
            int row = row0 + v + khalf * 8;
            U[(size_t)row * 192 + k * 64 + j * 16 + m] = acc[j][v];
        }
    }
}

// ---------------------------------------------------------------------------
// Phase 3: out[row][k*1024+e] = sum_rp U[row][k*64+rp] * A[k][e][rp]
// GEMM M=16384, N=1024 (per k), K=64. Each wave: 16 rows x 64 cols.
// ---------------------------------------------------------------------------
__global__ void __launch_bounds__(256)
k_phase3_UAt(const float* __restrict__ U, const float* __restrict__ A,
             float* __restrict__ out) {
    const int lane  = threadIdx.x & 31;
    const int wave  = threadIdx.x >> 5;
    const int m     = lane & 15;
    const int khalf = lane >> 4;
    const int row0  = blockIdx.x * 128 + wave * 16;
    const int n0    = blockIdx.y * 64;
    const int k     = blockIdx.z;

    const float* urow = U + (size_t)(row0 + m) * 192 + k * 64;
    const float* ak   = A + (size_t)k * 1024 * 64;          // A[k][e][rp]

    v8f acc[4] = {v8f{0}, v8f{0}, v8f{0}, v8f{0}};

#pragma unroll
    for (int kt = 0; kt < 16; ++kt) {
        const int kb = kt * 4 + khalf * 2;
        v2f a = *(const v2f*)(urow + kb);
#pragma unroll
        for (int j = 0; j < 4; ++j) {
            const int e = n0 + j * 16 + m;
            // B fragment = A^T[rp][e] = A[k][e][rp]: contiguous pair in rp
            v2f b = *(const v2f*)(ak + (size_t)e * 64 + kb);
            acc[j] = __builtin_amdgcn_wmma_f32_16x16x4_f32(
                false, a, false, b, (short)0, acc[j], false, false);
        }
    }

    // Streaming epilogue: 192 MB output, bypass caches with non-temporal stores
#pragma unroll
    for (int j = 0; j < 4; ++j) {
#pragma unroll
        for (int v = 0; v < 8; ++v) {
            int row = row0 + v + khalf * 8;
            float* p = out + (size_t)row * OUTW + k * 1024 + n0 + j * 16 + m;
            __builtin_nontemporal_store(acc[j][v], p);
        }
    }
}

// ---------------------------------------------------------------------------
// Launch
// ---------------------------------------------------------------------------
extern "C" void kernel_launch(void* const* d_in, const int* in_sizes, int n_in,
                              void* d_out, int out_size, void* d_ws, size_t ws_size,
                              hipStream_t stream) {
    const float* x  = (const float*)d_in[0];   // [2, 8192, 1024]
    const float* Ws = (const float*)d_in[1];   // [2304, 768]
    const float* A  = (const float*)d_in[2];   // [3, 1024, 64]
    const float* Bf = (const float*)d_in[3];   // [3, 64, 768]
    float* out = (float*)d_out;                // [2, 8192, 3072]

    char* ws = (char*)d_ws;
    float* U    = (float*)(ws);                         // 16384*192*4 = 12,582,912 B
    float* T    = (float*)(ws + 12582912);              // 3*64*768*4  =    589,824 B
    float* M    = (float*)(ws + 12582912 + 589824);     // 3*64*64*4   =     49,152 B
    float* A2T  = (float*)(ws + 12582912 + 589824 + 49152); // 3*64*1024*4 = 786,432 B

    // Tiny weight-space precompute (serial on stream; ~0.3 GFLOP total)
    k_precompute_T  <<<(3 * 64 * 768  + 255) / 256, 256, 0, stream>>>(Bf, Ws, T);
    k_precompute_M  <<<(3 * 64 * 64   + 255) / 256, 256, 0, stream>>>(T, Bf, M);
    k_precompute_A2T<<<(3 * 64 * 1024 + 255) / 256, 256, 0, stream>>>(A, M, A2T);

    // Phase 2: U = x @ (A M)   -> grid: 16384/128 row-blocks x 3 projections
    dim3 g2(NROWS / 128, 3, 1);
    k_phase2_xA2<<<g2, 256, 0, stream>>>(x, A2T, U);

    // Phase 3: y = U @ A^T     -> grid: row-blocks x 16 col-groups x 3
    dim3 g3(NROWS / 128, 1024 / 64, 3);
    k_phase3_UAt<<<g3, 256, 0, stream>>>(U, A, out);
}